// AvgPoolingModule_59322088292454
// MI455X (gfx1250) — compile-verified
//
#include <hip/hip_runtime.h>
#include <cstdint>

#define KGRID 64
#define KCELLS (KGRID * KGRID * KGRID) // 262144
#define CFEAT 64

typedef float4 f4;

// ---------------------------------------------------------------------------
// Kernel 1: zero d_out (sums/out) and counts with float4 stores.
// ---------------------------------------------------------------------------
__global__ __launch_bounds__(256) void voxmean_zero_kernel(float* __restrict__ out,
                                                           int n4out,
                                                           float* __restrict__ counts,
                                                           int n4cnt) {
    int i = blockIdx.x * blockDim.x + threadIdx.x;
    f4 z = make_float4(0.f, 0.f, 0.f, 0.f);
    if (i < n4out) {
        ((f4*)out)[i] = z;
    } else if (i - n4out < n4cnt) {
        ((f4*)counts)[i - n4out] = z;
    }
}

// ---------------------------------------------------------------------------
// Kernel 2: scatter-add. 256 threads/block = 8 waves; 16 points per tile
// (16 lanes x float4 = one 256B feature row per point). point_feat is
// streamed global->LDS with double-buffered async DMA (ASYNCcnt), so the
// 256MB feature stream is always one tile ahead of the atomic issue.
// ---------------------------------------------------------------------------
__global__ __launch_bounds__(256) void voxmean_scatter_kernel(
    const float* __restrict__ feat, const float* __restrict__ points,
    float* __restrict__ out, float* __restrict__ counts, int N, int numTiles) {
    __shared__ f4 smem[2][256]; // 2 buffers x 16 points x 16 float4 = 8 KB

    const int tid  = (int)threadIdx.x;
    const int lane = tid & 31;   // wave32
    const int sub  = lane & 15;  // lane within the 16-lane point group
    const f4* feat4 = (const f4*)feat;
    const long maxI4 = (long)N * 16 - 1;

    int t = (int)blockIdx.x;
    if (t >= numTiles) return;

    // Prologue: async-issue tile t into buffer 0 (this wave's own slice).
    {
        long gi = (long)t * 256 + tid;
        if (gi > maxI4) gi = maxI4;
        unsigned long long ga = (unsigned long long)(uintptr_t)(feat4 + gi);
        unsigned lds = (unsigned)(uintptr_t)&smem[0][tid];
        asm volatile("global_load_async_to_lds_b128 %0, %1, off"
                     :: "v"(lds), "v"(ga) : "memory");
    }

    int buf = 0;
    for (; t < numTiles; t += gridDim.x) {
        int tn = t + (int)gridDim.x;
        if (tn < numTiles) {
            // Prefetch next tile into the other buffer, then wait for current.
            long gi = (long)tn * 256 + tid;
            if (gi > maxI4) gi = maxI4;
            unsigned long long ga = (unsigned long long)(uintptr_t)(feat4 + gi);
            unsigned lds = (unsigned)(uintptr_t)&smem[buf ^ 1][tid];
            asm volatile("global_load_async_to_lds_b128 %0, %1, off"
                         :: "v"(lds), "v"(ga) : "memory");
            asm volatile("s_wait_asynccnt 0x1" ::: "memory");
        } else {
            asm volatile("s_wait_asynccnt 0x0" ::: "memory");
        }

        // Point handled by this 16-lane group.
        int point = t * 16 + ((tid >> 5) << 1) + (lane >> 4);

        // Leader lane computes the voxel cell index and bumps the count.
        int idx = 0;
        if (sub == 0 && point < N) {
            const float* pp = points + (size_t)point * 3;
            float x = pp[0], y = pp[1], z = pp[2];
            int ix = __float2int_rd((x + 1.0f) * (0.5f * KGRID));
            int iy = __float2int_rd((y + 1.0f) * (0.5f * KGRID));
            int iz = __float2int_rd((z + 1.0f) * (0.5f * KGRID));
            ix = min(max(ix, 0), KGRID - 1);
            iy = min(max(iy, 0), KGRID - 1);
            iz = min(max(iz, 0), KGRID - 1);
            idx = ix * (KGRID * KGRID) + iy * KGRID + iz;
            atomicAdd(&counts[idx], 1.0f);
        }
        // Broadcast idx from lane 0 / lane 16 within the wave (wave32).
        idx = __builtin_amdgcn_ds_bpermute((lane & 16) << 2, idx);

        // Read this lane's float4 of the feature row from LDS, scatter-add.
        f4 v = smem[buf][tid];
        if (point < N) {
            float* dst = out + (size_t)idx * CFEAT + sub * 4;
            atomicAdd(dst + 0, v.x);
            atomicAdd(dst + 1, v.y);
            atomicAdd(dst + 2, v.z);
            atomicAdd(dst + 3, v.w);
        }
        buf ^= 1;
    }
}

// ---------------------------------------------------------------------------
// Kernel 3: out[cell, c] = sums[cell, c] / max(count[cell], 1), float4-wide.
// ---------------------------------------------------------------------------
__global__ __launch_bounds__(256) void voxmean_normalize_kernel(
    float* __restrict__ out, const float* __restrict__ counts, int n4) {
    int i = blockIdx.x * blockDim.x + threadIdx.x;
    if (i >= n4) return;
    float c = counts[i >> 4]; // 16 float4 per 64-channel cell row
    float denom = (c > 0.0f) ? c : 1.0f;
    f4 v = ((f4*)out)[i];
    v.x /= denom;
    v.y /= denom;
    v.z /= denom;
    v.w /= denom;
    ((f4*)out)[i] = v;
}

// ---------------------------------------------------------------------------
extern "C" void kernel_launch(void* const* d_in, const int* in_sizes, int n_in,
                              void* d_out, int out_size, void* d_ws, size_t ws_size,
                              hipStream_t stream) {
    const float* feat   = (const float*)d_in[0]; // (N, 64) f32
    const float* points = (const float*)d_in[1]; // (N, 3)  f32
    float* out    = (float*)d_out;               // (64,64,64,64) f32
    float* counts = (float*)d_ws;                // 262144 f32 = 1 MB scratch

    const int N = in_sizes[0] / CFEAT;
    const int n4out = out_size / 4;  // 4,194,304 float4
    const int n4cnt = KCELLS / 4;    // 65,536 float4

    // 1) zero accumulators (harness poisons d_out / d_ws)
    int zn = n4out + n4cnt;
    voxmean_zero_kernel<<<(zn + 255) / 256, 256, 0, stream>>>(out, n4out, counts, n4cnt);

    // 2) scatter-add with async-to-LDS double-buffered feature streaming
    int numTiles = (N + 15) / 16; // 62,500 for N = 1e6
    int blocks = numTiles < 2048 ? numTiles : 2048;
    voxmean_scatter_kernel<<<blocks, 256, 0, stream>>>(feat, points, out, counts, N, numTiles);

    // 3) divide by per-cell counts
    voxmean_normalize_kernel<<<(n4out + 255) / 256, 256, 0, stream>>>(out, counts, n4out);
}